// HGT_240518168994
// MI455X (gfx1250) — compile-verified
//
#include <hip/hip_runtime.h>
#include <math.h>

typedef __attribute__((ext_vector_type(2))) float v2f;
typedef __attribute__((ext_vector_type(8))) float v8f;

#define N1V 10000
#define N2V 10000
#define DIN 128
#define HV 64
#define HEADSV 2
#define LV 2
#define EV 250000
#define PV 100000
#define HH 128   // HEADS*H

// ---------------- WMMA GEMM: out[M,Nc] = A[M,K] @ W[K,Nc] + bias, optional relu ----
// A row stride = lda, out row stride = ldOut. M%16==0, K%4==0, Nc%16==0, Nc<=128.
// One wave per 16x16 tile; 8 waves per block (256 threads) cover Nc columns.
__global__ void gemm_bias(const float* __restrict__ A, int lda,
                          const float* __restrict__ W,
                          const float* __restrict__ bias,
                          float* __restrict__ out, int ldOut,
                          int M, int K, int Nc, int doRelu)
{
    int wave = threadIdx.x >> 5;
    int lane = threadIdx.x & 31;
    if (wave * 16 >= Nc) return;              // wave-uniform: EXEC stays all-1s
    int row0 = blockIdx.x * 16;
    int col0 = wave * 16;
    int hi = lane >> 4;                       // 0/1
    int lo = lane & 15;

    v8f acc = {0.f, 0.f, 0.f, 0.f, 0.f, 0.f, 0.f, 0.f};
    const float* ap = A + (size_t)(row0 + lo) * lda + 2 * hi;
    const float* bp = W + (size_t)(2 * hi) * Nc + col0 + lo;
    for (int k0 = 0; k0 < K; k0 += 4) {
        v2f a, b;
        a[0] = ap[k0];
        a[1] = ap[k0 + 1];
        b[0] = bp[(size_t)k0 * Nc];
        b[1] = bp[(size_t)(k0 + 1) * Nc];
        acc = __builtin_amdgcn_wmma_f32_16x16x4_f32(
            false, a, false, b, (short)0, acc, false, false);
    }
    for (int r = 0; r < 8; ++r) {
        int m = row0 + r + 8 * hi;
        int n = col0 + lo;
        float v = acc[r] + bias[n];
        if (doRelu) v = fmaxf(v, 0.f);
        out[(size_t)m * ldOut + n] = v;
    }
}

// ---------------- Edge pass 1: alpha = leakyrelu(dot(k,q)/8), running global max ----
// One wave per edge: lane loads float4 (feats lane*4..lane*4+3); head = lane>>4.
__global__ void edge_alpha(const int* __restrict__ row, const int* __restrict__ col,
                           const float* __restrict__ kn, const float* __restrict__ qn,
                           float* __restrict__ alpha, unsigned int* __restrict__ maxKey,
                           int E)
{
    __shared__ float smax[2][8];
    int wave = threadIdx.x >> 5;
    int lane = threadIdx.x & 31;
    int e = blockIdx.x * 8 + wave;
    int valid = (e < E);
    float s = 0.f;
    if (valid) {
        int r = row[e], c = col[e];
        float4 k4 = ((const float4*)(kn + (size_t)r * HH))[lane];
        float4 q4 = ((const float4*)(qn + (size_t)c * HH))[lane];
        s = k4.x * q4.x + k4.y * q4.y + k4.z * q4.z + k4.w * q4.w;
    }
    // reduce within each 16-lane half (one head per half)
    for (int m = 8; m >= 1; m >>= 1) s += __shfl_xor(s, m, 32);
    if ((lane & 15) == 0) {
        int h = lane >> 4;
        float a = s * 0.125f;                 // 1/sqrt(64)
        a = (a > 0.f) ? a : 0.2f * a;         // leaky relu
        if (valid) {
            alpha[(size_t)e * 2 + h] = a;
            smax[h][wave] = a;
        } else {
            smax[h][wave] = -3.4e38f;
        }
    }
    __syncthreads();
    if (threadIdx.x < 2) {
        float m = smax[threadIdx.x][0];
        for (int i = 1; i < 8; ++i) m = fmaxf(m, smax[threadIdx.x][i]);
        unsigned b = __float_as_uint(m);      // monotonic order-preserving key
        unsigned key = (b & 0x80000000u) ? ~b : (b | 0x80000000u);
        atomicMax(&maxKey[threadIdx.x], key);
    }
}

// ---------------- Edge pass 2: alpha <- exp(alpha - max); accumulate per-head sums --
__global__ void edge_expsum(float* __restrict__ alpha,
                            const unsigned int* __restrict__ maxKey,
                            float* __restrict__ sums, int n)   // n = E*2
{
    __shared__ float ssum[2];
    if (threadIdx.x < 2) ssum[threadIdx.x] = 0.f;
    __syncthreads();
    int i = blockIdx.x * blockDim.x + threadIdx.x;
    if (i < n) {
        int h = i & 1;
        unsigned key = maxKey[h];
        unsigned b = (key & 0x80000000u) ? (key & 0x7FFFFFFFu) : ~key;
        float m = __uint_as_float(b);
        float e = __expf(alpha[i] - m);
        alpha[i] = e;
        atomicAdd(&ssum[h], e);
    }
    __syncthreads();
    if (threadIdx.x < 2) atomicAdd(&sums[threadIdx.x], ssum[threadIdx.x]);
}

// ---------------- Edge pass 3: agg[col] += v[row] * (exp/sum), per head -------------
// Thread handles one float4 chunk: idx over E*32.
__global__ void edge_scatter(const int* __restrict__ row, const int* __restrict__ col,
                             const float* __restrict__ vn, const float* __restrict__ alpha,
                             const float* __restrict__ sums, float* __restrict__ agg,
                             int E)
{
    int idx = blockIdx.x * blockDim.x + threadIdx.x;
    if (idx >= E * 32) return;
    int e = idx >> 5;
    int q = idx & 31;                         // float4 chunk -> feats q*4..q*4+3
    int h = q >> 4;                           // head
    float w = alpha[(size_t)e * 2 + h] / sums[h];
    int r = row[e], c = col[e];
    float4 v4 = ((const float4*)(vn + (size_t)r * HH))[q];
    float* dst = agg + (size_t)c * HH + q * 4;
    atomicAdd(dst + 0, v4.x * w);
    atomicAdd(dst + 1, v4.y * w);
    atomicAdd(dst + 2, v4.z * w);
    atomicAdd(dst + 3, v4.w * w);
}

// ---------------- Prediction: out[p] = dot(Em[a], Ed[b]) over 128 -------------------
__global__ void pred_kernel(const int* __restrict__ pa, const int* __restrict__ pb,
                            const float* __restrict__ Em, const float* __restrict__ Ed,
                            float* __restrict__ out, int P)
{
    int wave = threadIdx.x >> 5;
    int lane = threadIdx.x & 31;
    int p = blockIdx.x * 8 + wave;
    if (p >= P) return;
    const float* ea = Em + (size_t)pa[p] * HH;
    const float* eb = Ed + (size_t)pb[p] * HH;
    float s = 0.f;
    for (int j = 0; j < 4; ++j) s += ea[lane + 32 * j] * eb[lane + 32 * j];
    for (int m = 16; m >= 1; m >>= 1) s += __shfl_xor(s, m, 32);
    if (lane == 0) out[p] = s;
}

extern "C" void kernel_launch(void* const* d_in, const int* in_sizes, int n_in,
                              void* d_out, int out_size, void* d_ws, size_t ws_size,
                              hipStream_t stream)
{
    const float* x_n1 = (const float*)d_in[0];
    const float* x_n2 = (const float*)d_in[1];
    const int*   e12  = (const int*)d_in[2];      // [2,E]: rows(n1), cols(n2)
    const int*   e21  = (const int*)d_in[3];      // [2,E]: rows(n2), cols(n1)
    const int*   pe   = (const int*)d_in[4];      // [2,P]
    const float* Win  = (const float*)d_in[5];    // [2,128,64]
    const float* b_in = (const float*)d_in[6];    // [2,64]
    const float* Wk   = (const float*)d_in[7];    // [L,2,64,128]
    const float* bk   = (const float*)d_in[8];    // [L,2,128]
    const float* Wq   = (const float*)d_in[9];
    const float* bq   = (const float*)d_in[10];
    const float* Wv   = (const float*)d_in[11];
    const float* bv   = (const float*)d_in[12];
    const float* Wout = (const float*)d_in[13];   // [L,2,128,64]
    const float* bout = (const float*)d_in[14];   // [L,2,64]

    // ---- workspace carve-out (floats) ----
    float* ws = (float*)d_ws;
    size_t o = 0;
    float* x1   = ws + o; o += (size_t)N1V * HV;    // 640k
    float* x2   = ws + o; o += (size_t)N2V * HV;
    float* Em   = ws + o; o += (size_t)N1V * HH;    // feats1 concat [N1, L*H]
    float* Ed   = ws + o; o += (size_t)N2V * HH;
    float* kn   = ws + o; o += (size_t)N1V * HH;    // node-level K (10000x128)
    float* qn   = ws + o; o += (size_t)N1V * HH;
    float* vn   = ws + o; o += (size_t)N1V * HH;
    float* agg1 = ws + o; o += (size_t)N1V * HH;
    float* agg2 = ws + o; o += (size_t)N2V * HH;
    float* alpha = ws + o; o += (size_t)EV * 2;
    unsigned int* maxKey = (unsigned int*)(ws + o); o += 2;
    float* sums = ws + o; o += 2;

    // ---- input linears + relu ----
    gemm_bias<<<N1V / 16, 256, 0, stream>>>(x_n1, DIN, Win, b_in, x1, HV,
                                            N1V, DIN, HV, 1);
    gemm_bias<<<N2V / 16, 256, 0, stream>>>(x_n2, DIN, Win + DIN * HV, b_in + HV, x2, HV,
                                            N2V, DIN, HV, 1);

    for (int l = 0; l < LV; ++l) {
        const float* s1; int lda1;   // current n1 features
        const float* s2; int lda2;   // current n2 features
        if (l == 0) { s1 = x1; lda1 = HV; s2 = x2; lda2 = HV; }
        else        { s1 = Em + (l - 1) * HV; lda1 = HH;
                      s2 = Ed + (l - 1) * HV; lda2 = HH; }

        for (int t = 0; t < 2; ++t) {
            // t==0: n1 -> n2 over edge_12 ; t==1: n2 -> n1 over edge_21
            const float* src  = (t == 0) ? s1 : s2;  int ldas = (t == 0) ? lda1 : lda2;
            const float* dst  = (t == 0) ? s2 : s1;  int ldad = (t == 0) ? lda2 : lda1;
            const int*   row  = (t == 0) ? e12 : e21;
            const int*   colp = row + EV;
            float*       agg  = (t == 0) ? agg2 : agg1;
            int Nsrc = 10000, Ndst = 10000;
            size_t widx = (size_t)l * 2 + t;
            const float* Wk_ = Wk + widx * HV * HH;  const float* bk_ = bk + widx * HH;
            const float* Wq_ = Wq + widx * HV * HH;  const float* bq_ = bq + widx * HH;
            const float* Wv_ = Wv + widx * HV * HH;  const float* bv_ = bv + widx * HH;

            // node-level K/Q/V (gather commutes with linear)
            gemm_bias<<<Nsrc / 16, 256, 0, stream>>>(src, ldas, Wk_, bk_, kn, HH,
                                                     Nsrc, HV, HH, 0);
            gemm_bias<<<Ndst / 16, 256, 0, stream>>>(dst, ldad, Wq_, bq_, qn, HH,
                                                     Ndst, HV, HH, 0);
            gemm_bias<<<Nsrc / 16, 256, 0, stream>>>(src, ldas, Wv_, bv_, vn, HH,
                                                     Nsrc, HV, HH, 0);

            hipMemsetAsync(maxKey, 0, 2 * sizeof(unsigned int), stream); // key 0 == -inf
            hipMemsetAsync(sums, 0, 2 * sizeof(float), stream);
            hipMemsetAsync(agg, 0, (size_t)Ndst * HH * sizeof(float), stream);

            edge_alpha<<<(EV + 7) / 8, 256, 0, stream>>>(row, colp, kn, qn,
                                                         alpha, maxKey, EV);
            edge_expsum<<<(EV * 2 + 255) / 256, 256, 0, stream>>>(alpha, maxKey,
                                                                  sums, EV * 2);
            edge_scatter<<<(EV * 32 + 255) / 256, 256, 0, stream>>>(row, colp, vn,
                                                                    alpha, sums, agg, EV);
        }

        // output linears -> write into concat feature buffers at column l*H
        size_t w0 = (size_t)l * 2 + 0, w1 = (size_t)l * 2 + 1;
        gemm_bias<<<N1V / 16, 256, 0, stream>>>(agg1, HH, Wout + w0 * HH * HV,
                                                bout + w0 * HV, Em + l * HV, HH,
                                                N1V, HH, HV, 0);
        gemm_bias<<<N2V / 16, 256, 0, stream>>>(agg2, HH, Wout + w1 * HH * HV,
                                                bout + w1 * HV, Ed + l * HV, HH,
                                                N2V, HH, HV, 0);
    }

    // ---- prediction dot products ----
    pred_kernel<<<(PV + 7) / 8, 256, 0, stream>>>(pe, pe + PV, Em, Ed,
                                                  (float*)d_out, PV);
}